// BSCELossAdaptiveGra_66898410602953
// MI455X (gfx1250) — compile-verified
//
#include <hip/hip_runtime.h>
#include <hip/hip_bf16.h>

// ---------------------------------------------------------------------------
// BSCE adaptive-gamma focal loss, MI455X (gfx1250, wave32).
// Streaming single-pass softmax stats via async global->LDS copies (ASYNCcnt),
// double-buffered per wave; deterministic two-stage reduction.
// ---------------------------------------------------------------------------

#define WAVES_PER_BLOCK 8
#define ROWS_PER_WAVE   8
static constexpr int kC    = 1000;   // classes per row (reference shape)
static constexpr int kF4   = 250;    // float4 per row
static constexpr int kIter = 8;      // ceil(256/32) strided float4 iters/lane

// ---- CDNA5 async global->LDS copy (tracked by ASYNCcnt) -------------------
#if defined(__has_builtin)
#if __has_builtin(__builtin_amdgcn_global_load_async_to_lds_b128)
#define HAVE_ASYNC_BUILTIN 1
#endif
#if __has_builtin(__builtin_amdgcn_s_wait_asynccnt)
#define HAVE_WAITASYNC_BUILTIN 1
#endif
#endif

#ifdef HAVE_WAITASYNC_BUILTIN
#define WAIT_ASYNC(n) __builtin_amdgcn_s_wait_asynccnt(n)
#else
#define WAIT_ASYNC(n) asm volatile("s_wait_asynccnt %0" ::"n"(n) : "memory")
#endif

// Builtin signature (per hipcc diagnostic): typed v4i pointers, AS1 src / AS3 dst.
typedef int v4i __attribute__((vector_size(16)));
typedef __attribute__((address_space(1))) v4i* global_v4i_ptr;
typedef __attribute__((address_space(3))) v4i* lds_v4i_ptr;

__device__ __forceinline__ void async_copy_b128(const float4* gsrc, float4* ldst) {
#ifdef HAVE_ASYNC_BUILTIN
  __builtin_amdgcn_global_load_async_to_lds_b128(
      (global_v4i_ptr)gsrc, (lds_v4i_ptr)ldst,
      /*imm offset=*/0, /*cpol=*/0);
#else
  unsigned loff =
      (unsigned)(unsigned long long)(__attribute__((address_space(3))) char*)ldst;
  asm volatile("global_load_async_to_lds_b128 %0, %1, off"
               ::"v"(loff), "v"(gsrc)
               : "memory");
#endif
}

// Issue exactly 8 async b128 copies per wave for one row (250 float4).
__device__ __forceinline__ void prefetch_row(const float* __restrict__ grow,
                                             float4* lbuf, int lane) {
  const float4* g4 = (const float4*)grow;  // row offset = 4000 B -> 16B aligned
#pragma unroll
  for (int i = 0; i < kIter; ++i) {
    int idx = lane + 32 * i;
    if (idx < kF4) async_copy_b128(g4 + idx, lbuf + idx);
  }
}

// Compute -diff * logpt for one row whose logits sit in `lbuf` (LDS).
__device__ __forceinline__ float row_loss(const float4* lbuf,
                                          const float* __restrict__ x,
                                          const int* __restrict__ tgt,
                                          int row, int lane) {
  const float NEG = -3.0e38f;  // finite sentinel: exp underflows to 0, max-neutral
  float4 v[kIter];
#pragma unroll
  for (int i = 0; i < kIter; ++i) {
    int idx = lane + 32 * i;
    if (idx < kF4) v[i] = lbuf[idx];
    else           v[i] = make_float4(NEG, NEG, NEG, NEG);
  }
  // per-lane max over 32 values
  float m = NEG;
#pragma unroll
  for (int i = 0; i < kIter; ++i)
    m = fmaxf(m, fmaxf(fmaxf(v[i].x, v[i].y), fmaxf(v[i].z, v[i].w)));
  // per-lane sum of exp(x - m)
  float s = 0.0f;
#pragma unroll
  for (int i = 0; i < kIter; ++i)
    s += expf(v[i].x - m) + expf(v[i].y - m) + expf(v[i].z - m) + expf(v[i].w - m);

  // wave32 butterfly merge of (m, s) pairs
#pragma unroll
  for (int o = 16; o > 0; o >>= 1) {
    float mo = __shfl_xor(m, o, 32);
    float so = __shfl_xor(s, o, 32);
    float nm = fmaxf(m, mo);
    s = s * expf(m - nm) + so * expf(mo - nm);
    m = nm;
  }
  // all lanes now hold the row (max, sumexp); finish the scalar math
  int   t     = tgt[row];
  float xt    = x[(size_t)row * kC + t];
  float logpt = xt - m - logf(s);
  float p     = expf(logpt);
  float d     = 2.0f * (1.0f - p);           // == sum |one_hot - softmax|_1
  float d3    = d * d * d;
  float diff  = (p < 0.2f) ? d3 * d * d : d3;  // gamma 5 vs 3
  return -diff * logpt;
}

__global__ __launch_bounds__(256) void bsce_rows_kernel(
    const float* __restrict__ x, const int* __restrict__ tgt,
    float* __restrict__ partials, int N) {
  __shared__ float4 stage[WAVES_PER_BLOCK][2][kF4];  // 64000 B
  __shared__ float  wave_loss[WAVES_PER_BLOCK];

  const int wave = threadIdx.x >> 5;
  const int lane = threadIdx.x & 31;
  const int row0 = (blockIdx.x * WAVES_PER_BLOCK + wave) * ROWS_PER_WAVE;

  int rows = N - row0;
  if (rows > ROWS_PER_WAVE) rows = ROWS_PER_WAVE;
  if (rows < 0) rows = 0;

  float acc = 0.0f;
  if (rows > 0) prefetch_row(x + (size_t)row0 * kC, &stage[wave][0][0], lane);

  for (int r = 0; r < rows; ++r) {
    if (r + 1 < rows) {
      prefetch_row(x + (size_t)(row0 + r + 1) * kC, &stage[wave][(r + 1) & 1][0], lane);
      WAIT_ASYNC(8);   // 16 outstanding -> wait until only next row's 8 remain
    } else {
      WAIT_ASYNC(0);   // drain: current buffer complete
    }
    acc += row_loss(&stage[wave][r & 1][0], x, tgt, row0 + r, lane);
  }

  if (lane == 0) wave_loss[wave] = acc;
  __syncthreads();
  if (threadIdx.x == 0) {
    float b = 0.0f;
#pragma unroll
    for (int w = 0; w < WAVES_PER_BLOCK; ++w) b += wave_loss[w];
    partials[blockIdx.x] = b;
  }
}

__global__ __launch_bounds__(256) void bsce_reduce_kernel(
    const float* __restrict__ partials, float* __restrict__ out, int n) {
  __shared__ float sm[256];
  float a = 0.0f;
  for (int i = threadIdx.x; i < n; i += 256) a += partials[i];
  sm[threadIdx.x] = a;
  __syncthreads();
  for (int s = 128; s > 0; s >>= 1) {
    if ((int)threadIdx.x < s) sm[threadIdx.x] += sm[threadIdx.x + s];
    __syncthreads();
  }
  if (threadIdx.x == 0) out[0] = sm[0];
}

extern "C" void kernel_launch(void* const* d_in, const int* in_sizes, int n_in,
                              void* d_out, int out_size, void* d_ws, size_t ws_size,
                              hipStream_t stream) {
  const float* x   = (const float*)d_in[0];
  const int*   tgt = (const int*)d_in[1];
  float*       out = (float*)d_out;
  float*       partials = (float*)d_ws;

  const int N = in_sizes[1];  // 65536 rows
  const int rows_per_block = WAVES_PER_BLOCK * ROWS_PER_WAVE;  // 64
  const int blocks = (N + rows_per_block - 1) / rows_per_block;  // 1024

  bsce_rows_kernel<<<blocks, 256, 0, stream>>>(x, tgt, partials, N);
  bsce_reduce_kernel<<<1, 256, 0, stream>>>(partials, out, blocks);
}